// LightGCN_86131274154529
// MI455X (gfx1250) — compile-verified
//
#include <hip/hip_runtime.h>
#include <hip/hip_bf16.h>
#include <stdint.h>

#define TPB 256

typedef int v4i_t __attribute__((ext_vector_type(4)));

// gfx1250 async global->LDS (builtin existence + v4i* signature confirmed in rounds 1-2)
#if defined(__gfx1250__) && __has_builtin(__builtin_amdgcn_global_load_async_to_lds_b128)
#define HAVE_ASYNC_LDS 1
#endif

__device__ __forceinline__ void async_stage_b128(const float4* gp, float4* lp) {
#ifdef HAVE_ASYNC_LDS
    __builtin_amdgcn_global_load_async_to_lds_b128(
        (__attribute__((address_space(1))) v4i_t*)(float4*)gp,
        (__attribute__((address_space(3))) v4i_t*)lp, 0, 0);
#else
    *lp = *gp;
#endif
}

__device__ __forceinline__ void wait_async_le(int n) {
#if defined(__gfx1250__)
#if __has_builtin(__builtin_amdgcn_s_wait_asynccnt)
    if (n == 0) __builtin_amdgcn_s_wait_asynccnt(0);
    else        __builtin_amdgcn_s_wait_asynccnt(1);
#else
    if (n == 0) asm volatile("s_wait_asynccnt 0x0" ::: "memory");
    else        asm volatile("s_wait_asynccnt 0x1" ::: "memory");
#endif
    asm volatile("" ::: "memory");   // keep LDS reads after the wait
#endif
}

__global__ void zero_int_kernel(int* __restrict__ p, long n) {
    long i = (long)blockIdx.x * blockDim.x + threadIdx.x;
    long stride = (long)gridDim.x * blockDim.x;
    for (; i < n; i += stride) p[i] = 0;
}

// cnt[dst] += 1 over all directed edges
__global__ void count_kernel(const long long* __restrict__ dst,
                             int* __restrict__ cnt, long E) {
    long i = (long)blockIdx.x * blockDim.x + threadIdx.x;
    if (i < E) {
        __builtin_prefetch(dst + i + 4096, 0, 0);   // -> global_prefetch_b8 (speculative)
        atomicAdd(&cnt[(int)dst[i]], 1);
    }
}

// dinv[n] = deg>0 ? rsqrt(deg) : 0   (deg integer >=1 when >0, so max(deg,1) is moot)
__global__ void dinv_kernel(const int* __restrict__ cnt, float* __restrict__ dinv, long N) {
    long i = (long)blockIdx.x * blockDim.x + threadIdx.x;
    if (i < N) {
        int d = cnt[i];
        dinv[i] = (d > 0) ? rsqrtf((float)d) : 0.0f;
    }
}

// single-workgroup exclusive scan of cnt -> rowStart[0..N], cursor copy for the fill pass
__global__ void scan_kernel(const int* __restrict__ cnt, int* __restrict__ rowStart,
                            int* __restrict__ cursor, long N) {
    __shared__ int sh[TPB];
    __shared__ int carrySh;
    const int tid = threadIdx.x;
    if (tid == 0) carrySh = 0;
    __syncthreads();
    for (long base = 0; base < N; base += TPB) {
        long i = base + tid;
        int v = (i < N) ? cnt[i] : 0;
        sh[tid] = v;
        __syncthreads();
        for (int off = 1; off < TPB; off <<= 1) {   // Hillis-Steele inclusive scan
            int t = (tid >= off) ? sh[tid - off] : 0;
            __syncthreads();
            sh[tid] += t;
            __syncthreads();
        }
        int incl = sh[tid];
        int excl = incl - v;
        int carry = carrySh;
        if (i < N) { rowStart[i] = carry + excl; cursor[i] = carry + excl; }
        __syncthreads();                            // all reads of carrySh done
        if (tid == TPB - 1) carrySh = carry + incl;
        __syncthreads();
    }
    if (tid == 0) rowStart[N] = carrySh;            // == E
}

// counting-sort edges by dst: cols[] = src indices grouped per destination row
__global__ void fill_kernel(const long long* __restrict__ src, const long long* __restrict__ dst,
                            int* __restrict__ cursor, int* __restrict__ cols, long E) {
    long i = (long)blockIdx.x * blockDim.x + threadIdx.x;
    if (i < E) {
        __builtin_prefetch(src + i + 4096, 0, 0);
        int d = (int)dst[i];
        int p = atomicAdd(&cursor[d], 1);
        cols[p] = (int)src[i];
    }
}

// x0 = concat(eu, ei); acc (in d_out final slots) = x0 / 25
__global__ void init_kernel(const float* __restrict__ eu, const float* __restrict__ ei,
                            float* __restrict__ x0, float* __restrict__ out,
                            long U64, long N64) {
    long i = (long)blockIdx.x * blockDim.x + threadIdx.x;
    if (i < N64) {
        float v = (i < U64) ? eu[i] : ei[i - U64];
        x0[i] = v;
        const float s = 1.0f / 25.0f;
        if (i < U64) out[i]       = v * s;   // users_final region
        else         out[U64 + i] = v * s;   // items_final region: 2*U64 + (i-U64)
    }
}

// copy raw embeddings into output copy slots
__global__ void copy_kernel(const float* __restrict__ eu, const float* __restrict__ ei,
                            float* __restrict__ out, long U64, long I64) {
    long N64 = U64 + I64;
    long i = (long)blockIdx.x * blockDim.x + threadIdx.x;
    if (i < N64) {
        if (i < U64) out[U64 + i]       = eu[i];
        else         out[U64 + I64 + i] = ei[i - U64];
    }
}

// Pull-style atomic-free propagation: one node per 16-lane group (2 nodes/wave),
// lane q owns dim-quad q. Neighbor rows staged 2-deep via async global->LDS with a
// wave-uniform loop (both groups run to max degree, clamped addrs, masked accumulate).
// Epilogue fuses x_next store with acc += x_next/25 into d_out.
__global__ __launch_bounds__(TPB)
void gather_kernel(const int* __restrict__ rowStart, const int* __restrict__ cols,
                   const float* __restrict__ dinv, const float* __restrict__ xin,
                   float* __restrict__ xout, float* __restrict__ outAcc,
                   long U64, long N, long E)
{
    __shared__ float4 stage[2][TPB];
    const int tid  = threadIdx.x;
    const int lane = tid & 31;
    const int grp  = lane >> 4;            // which node within the wave
    const int q    = lane & 15;            // dim-quad
    const long n0  = (long)blockIdx.x * 16 + 2 * (tid >> 5);
    const long n   = n0 + grp;
    const bool nvalid = (n < N);

    const long m0 = (n0     < N) ? n0     : (N - 1);
    const long m1 = (n0 + 1 < N) ? n0 + 1 : (N - 1);
    const int r0 = rowStart[m0], z0 = rowStart[m0 + 1];
    const int r1 = rowStart[m1], z1 = rowStart[m1 + 1];
    const int d0 = z0 - r0, d1 = z1 - r1;
    const int degMax = (d0 > d1) ? d0 : d1;       // wave-uniform trip count

    const int myStart = (grp == 0) ? r0 : r1;
    int       myDeg   = (grp == 0) ? d0 : d1;
    if (!nvalid) myDeg = 0;

    float4 acc = make_float4(0.f, 0.f, 0.f, 0.f);
    int c0 = 0, c1 = 0;

    auto issue = [&](int j, int which) {
        int jj = (j < myDeg) ? j : (myDeg > 0 ? myDeg - 1 : 0);   // clamp: EXEC stays full
        long ci = (long)myStart + jj;
        if (ci > E - 1) ci = E - 1;                               // empty-row guard
        int c = cols[ci];                                         // broadcast within group
        if (which == 0) c0 = c; else c1 = c;
        const float4* gp = (const float4*)xin + ((long)c * 16 + q);  // 256B/neighbor coalesced
        async_stage_b128(gp, &stage[which][tid]);
    };
    auto consume = [&](int j, int which) {
        if (j < myDeg) {
            int c   = (which == 0) ? c0 : c1;
            float w = dinv[c];
            float4 v = stage[which][tid];
            acc.x += w * v.x; acc.y += w * v.y; acc.z += w * v.z; acc.w += w * v.w;
        }
    };

    if (degMax > 0) {
        issue(0, 0);
        int j = 0;
        while (true) {                                  // manually 2x unrolled: static buf index
            bool more = (j + 1 < degMax);
            if (more) issue(j + 1, 1);
            wait_async_le(more ? 1 : 0);                // in-order: oldest stage ready
            consume(j, 0);
            if (++j >= degMax) break;

            more = (j + 1 < degMax);
            if (more) issue(j + 1, 0);
            wait_async_le(more ? 1 : 0);
            consume(j, 1);
            if (++j >= degMax) break;
        }
    }

    if (nvalid) {
        float dn = dinv[n];                             // row norm hoisted out of the sum
        float4 r = make_float4(acc.x * dn, acc.y * dn, acc.z * dn, acc.w * dn);
        long idx = n * 64 + q * 4;
        *(float4*)(xout + idx) = r;
        const float s = 1.0f / 25.0f;
        float* ap = (idx < U64) ? (outAcc + idx) : (outAcc + U64 + idx);
        float4 a = *(float4*)ap;
        a.x += r.x * s; a.y += r.y * s; a.z += r.z * s; a.w += r.w * s;
        *(float4*)ap = a;
    }
}

extern "C" void kernel_launch(void* const* d_in, const int* in_sizes, int n_in,
                              void* d_out, int out_size, void* d_ws, size_t ws_size,
                              hipStream_t stream) {
    const long long* edge = (const long long*)d_in[0];   // int64, shape (2, E) flat
    const float*     eu   = (const float*)d_in[1];
    const float*     ei   = (const float*)d_in[2];

    const long E   = (long)in_sizes[0] / 2;
    const long U64 = (long)in_sizes[1];
    const long I64 = (long)in_sizes[2];
    const long N64 = U64 + I64;
    const long N   = N64 / 64;

    const long long* srcA = edge;
    const long long* dstA = edge + E;
    float* out = (float*)d_out;

    // workspace layout (4B units, 256B-aligned sections): ~95 MB total
    const long Npad = (N + 64) & ~63L;       // covers N+1 for rowStart
    const long Epad = (E + 63) & ~63L;
    int*   cnt      = (int*)d_ws;            // N
    int*   rowStart = cnt + Npad;            // N+1
    int*   cursor   = rowStart + Npad;       // N
    float* dinv     = (float*)(cursor + Npad);   // N
    int*   cols     = (int*)(dinv + Npad);   // E
    float* xA       = (float*)(cols + Epad); // N64
    float* xB       = xA + N64;              // N64

    const int gE   = (int)((E + TPB - 1) / TPB);
    const int gN   = (int)((N + TPB - 1) / TPB);
    const int gN64 = (int)((N64 + TPB - 1) / TPB);

    // CSR build (once; reused by all 4 layers)
    zero_int_kernel<<<512, TPB, 0, stream>>>(cnt, N);
    count_kernel<<<gE, TPB, 0, stream>>>(dstA, cnt, E);
    dinv_kernel<<<gN, TPB, 0, stream>>>(cnt, dinv, N);
    scan_kernel<<<1, TPB, 0, stream>>>(cnt, rowStart, cursor, N);
    fill_kernel<<<gE, TPB, 0, stream>>>(srcA, dstA, cursor, cols, E);

    init_kernel<<<gN64, TPB, 0, stream>>>(eu, ei, xA, out, U64, N64);

    float* cur = xA;
    float* nxt = xB;
    const int gGather = (int)((N + 15) / 16);
    for (int l = 0; l < 4; ++l) {
        gather_kernel<<<gGather, TPB, 0, stream>>>(rowStart, cols, dinv, cur, nxt,
                                                   out, U64, N, E);
        float* t = cur; cur = nxt; nxt = t;
    }
    copy_kernel<<<gN64, TPB, 0, stream>>>(eu, ei, out, U64, I64);
}